// waspGridSpatialIntegral_43662637531604
// MI455X (gfx1250) — compile-verified
//
#include <hip/hip_runtime.h>

typedef __attribute__((ext_vector_type(2))) float v2f;
typedef __attribute__((ext_vector_type(8))) float v8f;

#define IMG_W 512
#define TILES 32  // 512 / 16

// One wave per strip.
//  wid in [0, nx):    channel 0, cumsum along x. Strip = 16 rows x 512 cols.
//  wid in [nx, 2*nx): channel 1, cumsum along y. Strip = 512 rows x 16 cols.
// Scan of a 16x16 tile T:
//  x-scan: D = T * U  (U upper-triangular ones incl diag), T in A operand, U in B.
//  y-scan: D = L * T  (L lower-triangular ones incl diag), L in A operand, T in B.
// Carry seeded into the C accumulator, new carry extracted from last col/row of D.
__global__ __launch_bounds__(256) void wasp_scan_wmma_kernel(
    const float* __restrict__ in, float* __restrict__ out, int B) {
  const int lane  = threadIdx.x & 31;
  const int wid   = (blockIdx.x << 3) | (threadIdx.x >> 5);
  const int half  = lane >> 4;   // which 16-lane half
  const int l16   = lane & 15;
  const int kbase = half << 1;   // K lanes this half covers within a K=4 chunk

  const int nx = B * TILES;      // strips per channel
  if (wid >= 2 * nx) return;     // wave-uniform

  // Triangular operand, built in registers: value at (big=l16, small=kk) is
  // (kk <= l16). Same bits serve as U in the B operand (x-scan) and L in the
  // A operand (y-scan).
  v2f tri[4];
#pragma unroll
  for (int k = 0; k < 4; ++k) {
    const int kk = 4 * k + kbase;
    tri[k].x = (kk     <= l16) ? 1.0f : 0.0f;
    tri[k].y = (kk + 1 <= l16) ? 1.0f : 0.0f;
  }

  if (wid < nx) {
    // ---------------- channel 0: inclusive cumsum along x ----------------
    const int  b       = wid >> 5;
    const int  rowTile = wid & 31;
    const long base    = (long)b * 2 * IMG_W * IMG_W;  // channel 0 plane
    const float* src   = in + base;
    float*       dst   = out + base;
    const int    r0    = rowTile << 4;

    // carry[v] = running sum for row M = v + 8*half (uniform across N lanes)
    v8f carry = (v8f)0.0f;

    // Per-lane pointers. A operand: lane holds (M=l16, K=kbase..kbase+1).
    const float* rowPtr = src + (long)(r0 + l16) * IMG_W + kbase;
    float*       outPtr = dst + (long)(r0 + 8 * half) * IMG_W + l16;

    for (int t = 0; t < TILES; ++t) {
      const int colBase = t << 4;
      v2f a[4];
#pragma unroll
      for (int k = 0; k < 4; ++k)
        a[k] = *(const v2f*)(rowPtr + colBase + 4 * k);  // global_load_b64

      if (t + 1 < TILES)
        __builtin_prefetch(rowPtr + colBase + 16, 0, 0);

      v8f acc = carry;  // seed with carry -> D already includes prefix offset
#pragma unroll
      for (int k = 0; k < 4; ++k)
        acc = __builtin_amdgcn_wmma_f32_16x16x4_f32(
            false, a[k], false, tri[k], (short)0, acc, false, false);

      // Store D: VGPR v -> row r0 + v + 8*half, col colBase + l16 (2x64B/VGPR)
#pragma unroll
      for (int v = 0; v < 8; ++v)
        outPtr[(long)v * IMG_W + colBase] = acc[v];

      // New carry = column N=15 of D, broadcast within each 16-lane half.
      const int srcLane = (lane & 16) | 15;
#pragma unroll
      for (int v = 0; v < 8; ++v)
        carry[v] = __shfl(acc[v], srcLane, 32);
    }
  } else {
    // ---------------- channel 1: inclusive cumsum along y ----------------
    const int  s       = wid - nx;
    const int  b       = s >> 5;
    const int  colTile = s & 31;
    const long base    = (long)b * 2 * IMG_W * IMG_W + (long)IMG_W * IMG_W;
    const float* src   = in + base;
    float*       dst   = out + base;
    const int    c0    = colTile << 4;

    float carry = 0.0f;  // running sum for column N = c0 + l16

    // B operand: lane holds (K = rowBase + 4k + kbase + {0,1}, N = l16).
    const float* colPtr = src + c0 + l16 + (long)kbase * IMG_W;
    float*       outPtr = dst + c0 + l16 + (long)(8 * half) * IMG_W;

    for (int t = 0; t < TILES; ++t) {
      const int rowBase = t << 4;
      v2f bd[4];
#pragma unroll
      for (int k = 0; k < 4; ++k) {
        const float* p = colPtr + (long)(rowBase + 4 * k) * IMG_W;
        bd[k].x = p[0];      // 64B coalesced across lanes 0-15 / 16-31
        bd[k].y = p[IMG_W];
      }

      if (t + 1 < TILES)
        __builtin_prefetch(colPtr + (long)(rowBase + 16) * IMG_W, 0, 0);

      v8f acc;
#pragma unroll
      for (int v = 0; v < 8; ++v) acc[v] = carry;  // seed all rows with carry
#pragma unroll
      for (int k = 0; k < 4; ++k)
        acc = __builtin_amdgcn_wmma_f32_16x16x4_f32(
            false, tri[k], false, bd[k], (short)0, acc, false, false);

      // Store D: VGPR v -> row rowBase + v + 8*half, col c0 + l16
#pragma unroll
      for (int v = 0; v < 8; ++v)
        outPtr[(long)(rowBase + v) * IMG_W] = acc[v];

      // New carry = row M=15 of D = VGPR 7, upper half; fetch per-column value.
      carry = __shfl(acc[7], 16 + l16, 32);
    }
  }
}

extern "C" void kernel_launch(void* const* d_in, const int* in_sizes, int n_in,
                              void* d_out, int out_size, void* d_ws, size_t ws_size,
                              hipStream_t stream) {
  const float* in  = (const float*)d_in[0];
  float*       out = (float*)d_out;
  const int B = in_sizes[0] / (2 * IMG_W * IMG_W);  // 128 for the reference
  const int waves  = 2 * B * TILES;                 // 8192
  const int blocks = (waves + 7) / 8;               // 8 waves (256 thr) / block
  wasp_scan_wmma_kernel<<<blocks, 256, 0, stream>>>(in, out, B);
}